// TransitionUp_23536420782445
// MI455X (gfx1250) — compile-verified
//
#include <hip/hip_runtime.h>

#define C 256
#define S 16
#define NPTS 262144
#define BSTRIDE 272   // padded LDS stride for B (bank-conflict-free half-wave rows)
#define ASTRIDE 260   // padded LDS stride for A

typedef float v2f __attribute__((ext_vector_type(2)));
typedef float v8f __attribute__((ext_vector_type(8)));

// ---------------- Kernel 1: segment partial sums (atomic flush per overlap) ----
__global__ void seg_sum_kernel(const float* __restrict__ x,
                               const long long* __restrict__ o,
                               float* __restrict__ sums) {
  const int tid = threadIdx.x;              // channel
  const int r0 = blockIdx.x * 1024;
  const int r1 = r0 + 1024;
  int st = 0;
  for (int s = 0; s < S; ++s) {
    int en = (int)o[s];
    int lo = r0 > st ? r0 : st;
    int hi = r1 < en ? r1 : en;
    if (lo < hi) {
      float acc = 0.f;
      for (int r = lo; r < hi; ++r) acc += x[(size_t)r * C + tid];
      atomicAdd(&sums[s * C + tid], acc);
    }
    st = en;
  }
}

// ---------------- Kernel 2: mean -> h -> fused per-segment shift + BN scale ----
__global__ void seg_mlp_kernel(const long long* __restrict__ o,
                               const float* __restrict__ sums,
                               const float* __restrict__ W2,
                               const float* __restrict__ b2,
                               const float* __restrict__ W1,
                               const float* __restrict__ b1,
                               const float* __restrict__ gamma,
                               const float* __restrict__ beta,
                               const float* __restrict__ rmean,
                               const float* __restrict__ rvar,
                               float* __restrict__ shift,
                               float* __restrict__ scale_out) {
  __shared__ float smean[C];
  __shared__ float sh[C];
  const int s = blockIdx.x;
  const int tid = threadIdx.x;
  const int st = (s == 0) ? 0 : (int)o[s - 1];
  const int en = (int)o[s];
  const float inv_cnt = 1.0f / (float)(en - st);
  smean[tid] = sums[s * C + tid] * inv_cnt;
  __syncthreads();
  float acc = b2[tid];
  for (int k = 0; k < C; ++k) acc = fmaf(smean[k], W2[k * C + tid], acc);
  sh[tid] = fmaxf(acc, 0.f);
  __syncthreads();
  float acc2 = b1[tid];
  for (int k = 0; k < C; ++k) acc2 = fmaf(sh[k], W1[(C + k) * C + tid], acc2);
  const float sc = gamma[tid] * rsqrtf(rvar[tid] + 1e-5f);
  shift[s * C + tid] = (acc2 - rmean[tid]) * sc + beta[tid];
  if (s == 0) scale_out[tid] = sc;
}

// ---------------- Kernel 3: y = relu((x @ W1_top) * scale + shift[seg]) -------
// Block: 256 threads = 8 waves. Block tile: 32 rows x 256 cols.
// Wave tile: 16 rows x 64 cols (4 x v8f accumulators), WMMA f32 16x16x4.
// LDS: full W1_top (256x256, stride 272) + 32-row x tile (stride 260).
__global__ void __launch_bounds__(256)
gemm_fused_kernel(const float* __restrict__ x,
                  const float* __restrict__ W1,
                  const long long* __restrict__ o,
                  const float* __restrict__ scale,
                  const float* __restrict__ shift,
                  float* __restrict__ out) {
  extern __shared__ float lds[];
  float* ldsB = lds;                    // [256][BSTRIDE]
  float* ldsA = lds + C * BSTRIDE;      // [32][ASTRIDE]
  const int tid = threadIdx.x;
  const int m_base = blockIdx.x * 32;

  // Stage B (entire 256x256 W1_top) and A (32 rows of x), float4 coalesced.
  {
    const int r  = tid >> 6;            // 0..3
    const int c4 = (tid & 63) << 2;     // 0..252
    for (int k0 = 0; k0 < C; k0 += 4) {
      const int row = k0 + r;
      *(float4*)&ldsB[row * BSTRIDE + c4] = *(const float4*)&W1[row * C + c4];
    }
    for (int r0 = 0; r0 < 32; r0 += 4) {
      const int row = r0 + r;
      *(float4*)&ldsA[row * ASTRIDE + c4] =
          *(const float4*)&x[(size_t)(m_base + row) * C + c4];
    }
  }
  __syncthreads();

  const int wave = tid >> 5;
  const int lane = tid & 31;
  const int half = lane >> 4;           // half-wave id
  const int lr   = lane & 15;
  const int mloc = (wave >> 2) << 4;    // 0 or 16 (local row base)
  const int n0   = (wave & 3) << 6;     // 0,64,128,192 (col group)

  v8f acc[4] = {};
  const float* arow = &ldsA[(mloc + lr) * ASTRIDE + 2 * half];

  for (int k = 0; k < C; k += 4) {
    v2f a;
    a.x = arow[k];                      // A[M=lr][k + 2*half]
    a.y = arow[k + 1];                  // A[M=lr][k + 2*half + 1]
    const float* b0 = &ldsB[(k + 2 * half) * BSTRIDE + n0 + lr];
    const float* b1r = b0 + BSTRIDE;    // row k + 1 + 2*half
#pragma unroll
    for (int t = 0; t < 4; ++t) {
      v2f b;
      b.x = b0[16 * t];                 // B[k+2*half][n]
      b.y = b1r[16 * t];                // B[k+1+2*half][n]
      acc[t] = __builtin_amdgcn_wmma_f32_16x16x4_f32(
          false, a, false, b, (short)0, acc[t], false, false);
    }
  }

  // Epilogue: per-row segment id (incremental searchsorted over 16 offsets),
  // then relu(acc*scale[col] + shift[seg][col]).
  int oi[S];
#pragma unroll
  for (int j = 0; j < S; ++j) oi[j] = (int)o[j];
  const int row_base = m_base + mloc + 8 * half;  // D layout: VGPR r -> M=r+8*half
  int segr[8];
  {
    int sg = 0;
#pragma unroll
    for (int r = 0; r < 8; ++r) {
      const int row = row_base + r;
      while (sg < S - 1 && oi[sg] <= row) ++sg;
      segr[r] = sg;
    }
  }
#pragma unroll
  for (int t = 0; t < 4; ++t) {
    const int col = n0 + 16 * t + lr;
    const float sc = scale[col];
#pragma unroll
    for (int r = 0; r < 8; ++r) {
      const float v = acc[t][r] * sc + shift[segr[r] * C + col];
      out[(size_t)(row_base + r) * C + col] = fmaxf(v, 0.f);
    }
  }
}

// ---------------- Host launcher ------------------------------------------------
extern "C" void kernel_launch(void* const* d_in, const int* in_sizes, int n_in,
                              void* d_out, int out_size, void* d_ws, size_t ws_size,
                              hipStream_t stream) {
  (void)in_sizes; (void)n_in; (void)out_size; (void)ws_size;
  const float*     x     = (const float*)d_in[0];
  const long long* o     = (const long long*)d_in[1];
  const float*     W2    = (const float*)d_in[2];
  const float*     b2    = (const float*)d_in[3];
  const float*     W1    = (const float*)d_in[4];
  const float*     b1    = (const float*)d_in[5];
  const float*     gamma = (const float*)d_in[6];
  const float*     beta  = (const float*)d_in[7];
  const float*     rmean = (const float*)d_in[8];
  const float*     rvar  = (const float*)d_in[9];
  float* out = (float*)d_out;

  float* sums  = (float*)d_ws;          // [S*C]
  float* shift = sums + S * C;          // [S*C]
  float* scale = shift + S * C;         // [C]

  hipMemsetAsync(sums, 0, S * C * sizeof(float), stream);
  seg_sum_kernel<<<NPTS / 1024, 256, 0, stream>>>(x, o, sums);
  seg_mlp_kernel<<<S, C, 0, stream>>>(o, sums, W2, b2, W1, b1, gamma, beta,
                                      rmean, rvar, shift, scale);

  const size_t shmem = (size_t)(C * BSTRIDE + 32 * ASTRIDE) * sizeof(float); // 311808 B
  hipFuncSetAttribute(reinterpret_cast<const void*>(gemm_fused_kernel),
                      hipFuncAttributeMaxDynamicSharedMemorySize, (int)shmem);
  gemm_fused_kernel<<<NPTS / 32, 256, shmem, stream>>>(x, W1, o, scale, shift, out);
}